// GraphAttentionLayer_18588618457638
// MI455X (gfx1250) — compile-verified
//
#include <hip/hip_runtime.h>
#include <math.h>

typedef float v2f __attribute__((ext_vector_type(2)));
typedef float v8f __attribute__((ext_vector_type(8)));

#define N_NODES 4096
#define IN_F    512
#define NHEADS  8
#define OUT_F   64
#define HID     (NHEADS * OUT_F)   // 512
#define GAT_ALPHA 0.2f

// ---------------------------------------------------------------------------
// Kernel 1: wh = x @ weight   (fp32, V_WMMA_F32_16X16X4_F32)
// Block: 256 threads = 8 waves -> 32(M) x 64(N) tile. Grid: (4096/32, 512/64).
// ---------------------------------------------------------------------------
__global__ __launch_bounds__(256) void gat_gemm_wh(
    const float* __restrict__ x, const float* __restrict__ w,
    float* __restrict__ wh)
{
    __shared__ float As[32][17];   // 32 M-rows x 16 K, padded (bank conflicts)
    __shared__ float Bs[16][64];   // 16 K-rows x 64 N-cols

    const int tid  = threadIdx.x;
    const int lane = tid & 31;
    const int wave = tid >> 5;
    const int r    = lane & 15;     // M (for A/D) or N (for B/D) index
    const int half = lane >> 4;     // selects K pair (A/B) or M+8 (C/D)
    const int wm   = wave >> 2;     // 0..1  -> M sub-tile
    const int wn   = wave & 3;      // 0..3  -> N sub-tile
    const int m0   = blockIdx.x * 32;
    const int n0   = blockIdx.y * 64;

    v8f acc = {};

    for (int k = 0; k < IN_F; k += 16) {
        // Stage A tile (512 elems, 2/thread, 16-wide contiguous runs)
        {
            const int e0 = tid, e1 = tid + 256;
            As[e0 >> 4][e0 & 15] = x[(m0 + (e0 >> 4)) * IN_F + k + (e0 & 15)];
            As[e1 >> 4][e1 & 15] = x[(m0 + (e1 >> 4)) * IN_F + k + (e1 & 15)];
        }
        // Stage B tile (1024 elems, 4/thread, 64-wide contiguous runs)
        #pragma unroll
        for (int q = 0; q < 4; ++q) {
            const int e = tid + 256 * q;
            Bs[e >> 6][e & 63] = w[(k + (e >> 6)) * HID + n0 + (e & 63)];
        }
        __syncthreads();

        #pragma unroll
        for (int g = 0; g < 4; ++g) {
            const int kk = 4 * g + 2 * half;
            v2f a, b;
            a.x = As[wm * 16 + r][kk];
            a.y = As[wm * 16 + r][kk + 1];
            b.x = Bs[kk][wn * 16 + r];
            b.y = Bs[kk + 1][wn * 16 + r];
            acc = __builtin_amdgcn_wmma_f32_16x16x4_f32(
                false, a, false, b, (short)0, acc, false, false);
        }
        __syncthreads();
    }

    // D layout: VGPR v -> M = v + 8*half, N = r
    #pragma unroll
    for (int v = 0; v < 8; ++v)
        wh[(size_t)(m0 + wm * 16 + half * 8 + v) * HID + n0 + wn * 16 + r] = acc[v];
}

// ---------------------------------------------------------------------------
// Kernel 2: awh_i[n,H], awh_j[n,H]  — one wave per (node, head), 64-elem dot.
// ---------------------------------------------------------------------------
__global__ __launch_bounds__(256) void gat_scores(
    const float* __restrict__ wh, const float* __restrict__ li,
    const float* __restrict__ lj, float* __restrict__ awh_i,
    float* __restrict__ awh_j)
{
    const int i    = blockIdx.x;
    const int lane = threadIdx.x & 31;
    const int h    = threadIdx.x >> 5;          // 8 waves = 8 heads

    const float* row = wh + (size_t)i * HID + h * OUT_F;
    const float v0 = row[lane];
    const float v1 = row[lane + 32];
    float si = v0 * li[h * OUT_F + lane] + v1 * li[h * OUT_F + lane + 32];
    float sj = v0 * lj[h * OUT_F + lane] + v1 * lj[h * OUT_F + lane + 32];

    #pragma unroll
    for (int off = 16; off > 0; off >>= 1) {
        si += __shfl_down(si, off, 32);
        sj += __shfl_down(sj, off, 32);
    }
    if (lane == 0) {
        awh_i[i * NHEADS + h] = si;
        awh_j[i * NHEADS + h] = sj;
    }
}

// ---------------------------------------------------------------------------
// Kernel 3: fused mask + softmax + aggregation. One block per destination
// node i. Phase 1: coalesced scan of adj row -> LDS neighbor list (~1% dense).
// Phase 2: wave h runs online softmax over neighbors; masked entries would
// contribute exp(x - 1e10) == 0 in fp32, so the sparse sum is exact.
// ---------------------------------------------------------------------------
__global__ __launch_bounds__(256) void gat_attention(
    const float* __restrict__ adj, const float* __restrict__ wh,
    const float* __restrict__ awh_i, const float* __restrict__ awh_j,
    float* __restrict__ out)
{
    __shared__ int nbr[N_NODES];
    __shared__ int cnt;

    const int i   = blockIdx.x;
    const int tid = threadIdx.x;
    if (tid == 0) cnt = 0;
    __syncthreads();

    const float* arow = adj + (size_t)i * N_NODES;
    for (int j = tid; j < N_NODES; j += 256) {
        if (arow[j] != 0.0f) {
            const int p = atomicAdd(&cnt, 1);
            nbr[p] = j;
        }
    }
    __syncthreads();

    const int h    = tid >> 5;      // wave -> head
    const int lane = tid & 31;      // lane -> feature pair (2 floats)
    const int n    = cnt;
    const float ei = awh_i[i * NHEADS + h];

    float m = -INFINITY, l = 0.f, acc0 = 0.f, acc1 = 0.f;
    for (int p = 0; p < n; ++p) {
        const int j  = nbr[p];
        float e = ei + awh_j[j * NHEADS + h];
        e = (e >= 0.f) ? e : GAT_ALPHA * e;           // leaky-relu
        const float mn    = fmaxf(m, e);
        const float scale = __expf(m - mn);           // exp(-inf)=0 first iter
        const float pexp  = __expf(e - mn);
        const float2 whj  = *(const float2*)(wh + (size_t)j * HID +
                                             h * OUT_F + 2 * lane);
        l    = l    * scale + pexp;
        acc0 = acc0 * scale + pexp * whj.x;
        acc1 = acc1 * scale + pexp * whj.y;
        m = mn;
    }

    const float inv = (n > 0) ? (1.0f / l) : 0.0f;
    float2 o;
    o.x = acc0 * inv;
    o.y = acc1 * inv;
    *(float2*)(out + (size_t)i * HID + h * OUT_F + 2 * lane) = o;
}

// ---------------------------------------------------------------------------
extern "C" void kernel_launch(void* const* d_in, const int* in_sizes, int n_in,
                              void* d_out, int out_size, void* d_ws, size_t ws_size,
                              hipStream_t stream) {
    (void)in_sizes; (void)n_in; (void)out_size; (void)ws_size;
    const float* x   = (const float*)d_in[0];   // [4096, 512]
    const float* adj = (const float*)d_in[1];   // [4096, 4096]
    const float* w   = (const float*)d_in[2];   // [512, 512]
    const float* li  = (const float*)d_in[3];   // [1, 8, 64]
    const float* lj  = (const float*)d_in[4];   // [1, 8, 64]
    float* out = (float*)d_out;                 // [4096, 512]

    float* wh    = (float*)d_ws;                        // 4096*512 fp32 = 8 MB
    float* awh_i = wh + (size_t)N_NODES * HID;          // 4096*8
    float* awh_j = awh_i + (size_t)N_NODES * NHEADS;    // 4096*8

    dim3 g1(N_NODES / 32, HID / 64);
    gat_gemm_wh<<<g1, 256, 0, stream>>>(x, w, wh);
    gat_scores<<<N_NODES, 256, 0, stream>>>(wh, li, lj, awh_i, awh_j);
    gat_attention<<<N_NODES, 256, 0, stream>>>(adj, wh, awh_i, awh_j, out);
}